// GraphSageClassification_84576495993386
// MI455X (gfx1250) — compile-verified
//
#include <hip/hip_runtime.h>
#include <hip/hip_bf16.h>

typedef __attribute__((ext_vector_type(2))) float v2f;
typedef __attribute__((ext_vector_type(8))) float v8f;
typedef int gs_v4i __attribute__((vector_size(16)));   // matches builtin param pointee

#define FDIM 128
#define LDA  132   // padded LDS row stride: 132 % 64 == 4 -> conflict-free fragment reads

#define GS_AS1 __attribute__((address_space(1)))
#define GS_AS3 __attribute__((address_space(3)))

#if __has_builtin(__builtin_amdgcn_global_load_async_to_lds_b128) && \
    __has_builtin(__builtin_amdgcn_s_wait_asynccnt)
#define GS_HAS_ASYNC_LDS 1
#else
#define GS_HAS_ASYNC_LDS 0
#endif

// ---------------- zero scratch ----------------
__global__ __launch_bounds__(256)
void gs_zero_kernel(float* __restrict__ p, int n) {
    int i = blockIdx.x * blockDim.x + threadIdx.x;
    int stride = gridDim.x * blockDim.x;
    for (; i < n; i += stride) p[i] = 0.0f;
}

// ---------------- scatter-mean numerator/denominator ----------------
// one wave per edge: 32 lanes x float4 = 512B row; accumulators stay L2-resident
__global__ __launch_bounds__(256)
void gs_scatter_kernel(const float* __restrict__ x,
                       const int* __restrict__ src,
                       const int* __restrict__ dst,
                       float* __restrict__ sum,
                       float* __restrict__ cnt,
                       int E) {
    int gw   = (blockIdx.x * blockDim.x + threadIdx.x) >> 5;
    int lane = threadIdx.x & 31;
    int nw   = (gridDim.x * blockDim.x) >> 5;
    for (int e = gw; e < E; e += nw) {
        int s = src[e];
        int d = dst[e];
        const float4 v = ((const float4*)(x + (size_t)s * FDIM))[lane];
        float* ds = sum + (size_t)d * FDIM + lane * 4;
        __hip_atomic_fetch_add(ds + 0, v.x, __ATOMIC_RELAXED, __HIP_MEMORY_SCOPE_AGENT);
        __hip_atomic_fetch_add(ds + 1, v.y, __ATOMIC_RELAXED, __HIP_MEMORY_SCOPE_AGENT);
        __hip_atomic_fetch_add(ds + 2, v.z, __ATOMIC_RELAXED, __HIP_MEMORY_SCOPE_AGENT);
        __hip_atomic_fetch_add(ds + 3, v.w, __ATOMIC_RELAXED, __HIP_MEMORY_SCOPE_AGENT);
        if (lane == 0)
            __hip_atomic_fetch_add(cnt + d, 1.0f, __ATOMIC_RELAXED, __HIP_MEMORY_SCOPE_AGENT);
    }
}

// ---------------- fused SAGEConv GEMM (fp32 WMMA, LDS-staged A tiles) ----------------
// out[m][n] = act( (sum[m]/max(cnt[m],1)) . Wl[n] + bias[n] + x[m] . Wr[n] )
// block = 256 threads = 8 waves; wave w owns output-feature tile n0 = 16*w;
// block b owns node tile m0 = 16*b.  N must be a multiple of 16 (50000 = 3125*16).
__global__ __launch_bounds__(256)
void gs_sage_gemm_kernel(const float* __restrict__ sum,
                         const float* __restrict__ cnt,
                         const float* __restrict__ x,
                         const float* __restrict__ Wl,
                         const float* __restrict__ bias,
                         const float* __restrict__ Wr,
                         float* __restrict__ out,
                         int applyAct) {
    __shared__ float sA[16 * LDA];   // aggregate (un-normalized sum) tile
    __shared__ float sX[16 * LDA];   // root-feature tile

    const int tid  = threadIdx.x;
    const int lane = tid & 31;
    const int wave = tid >> 5;
    const int row  = lane & 15;
    const int hi   = lane >> 4;
    const int m0   = blockIdx.x * 16;
    const int n0   = wave * 16;

    // ---- stage both 16x128 A tiles into LDS (async global->LDS if available) ----
    const float* gA = sum + (size_t)m0 * FDIM;
    const float* gX = x   + (size_t)m0 * FDIM;
#if GS_HAS_ASYNC_LDS
    {
#pragma unroll
        for (int j = tid; j < 512; j += 256) {     // 512 x 16B per matrix
            int r = j >> 5, seg = j & 31;
            __builtin_amdgcn_global_load_async_to_lds_b128(
                (GS_AS1 gs_v4i*)(gA + (size_t)r * FDIM + seg * 4),
                (GS_AS3 gs_v4i*)(sA + r * LDA + seg * 4), 0, 0);
            __builtin_amdgcn_global_load_async_to_lds_b128(
                (GS_AS1 gs_v4i*)(gX + (size_t)r * FDIM + seg * 4),
                (GS_AS3 gs_v4i*)(sX + r * LDA + seg * 4), 0, 0);
        }
        __builtin_amdgcn_s_wait_asynccnt(0);
    }
#else
#pragma unroll
    for (int j = tid; j < 512; j += 256) {
        int r = j >> 5, seg = j & 31;
        *(float4*)(sA + r * LDA + seg * 4) = *(const float4*)(gA + (size_t)r * FDIM + seg * 4);
        *(float4*)(sX + r * LDA + seg * 4) = *(const float4*)(gX + (size_t)r * FDIM + seg * 4);
    }
#endif
    __syncthreads();

    const float inv = 1.0f / fmaxf(cnt[m0 + row], 1.0f);

    // A fragment addressing (16x4 f32 tile): lane row = lane&15, K pair = 2*hi
    const float* lA = sA + row * LDA + 2 * hi;
    const float* lX = sX + row * LDA + 2 * hi;
    // B fragment: B[k][n] = W[n][k]  (out = A @ W^T); weights stay in WGP$/L2
    const float* Brow = Wl + (size_t)(n0 + row) * FDIM + 2 * hi;
    const float* Crow = Wr + (size_t)(n0 + row) * FDIM + 2 * hi;

    __builtin_prefetch(Crow, 0, 0);   // global_prefetch_b8: pull lin_r weight row early

    v8f acc = {0.f, 0.f, 0.f, 0.f, 0.f, 0.f, 0.f, 0.f};

#pragma unroll
    for (int k = 0; k < FDIM; k += 4) {
        v2f a = { lA[k] * inv, lA[k + 1] * inv };
        v2f b = { Brow[k],     Brow[k + 1] };
        acc = __builtin_amdgcn_wmma_f32_16x16x4_f32(false, a, false, b,
                                                    (short)0, acc, false, false);
    }
#pragma unroll
    for (int k = 0; k < FDIM; k += 4) {
        v2f a = { lX[k], lX[k + 1] };
        v2f b = { Crow[k], Crow[k + 1] };
        acc = __builtin_amdgcn_wmma_f32_16x16x4_f32(false, a, false, b,
                                                    (short)0, acc, false, false);
    }

    // D layout: lane col = lane&15, VGPR r -> node m = r + 8*hi
    const float bv = bias[n0 + row];
    float* orow = out + (size_t)m0 * FDIM + n0 + row;
#pragma unroll
    for (int r = 0; r < 8; ++r) {
        int m = r + 8 * hi;
        float v = acc[r] + bv;
        if (applyAct) v = fmaxf(v, 0.2f * v);   // leaky_relu(0.2), branchless
        orow[(size_t)m * FDIM] = v;
    }
}

// ---------------- classifier: logits = y2 @ Wc^T + bc  (C = 2) ----------------
__global__ __launch_bounds__(256)
void gs_classifier_kernel(const float* __restrict__ y2,
                          const float* __restrict__ Wc,
                          const float* __restrict__ bc,
                          float* __restrict__ logits,
                          int n) {
    int gw   = (blockIdx.x * blockDim.x + threadIdx.x) >> 5;
    int lane = threadIdx.x & 31;
    int nw   = (gridDim.x * blockDim.x) >> 5;
    const float4 w0 = ((const float4*)Wc)[lane];
    const float4 w1 = ((const float4*)(Wc + FDIM))[lane];
    for (int node = gw; node < n; node += nw) {
        const float4 v = ((const float4*)(y2 + (size_t)node * FDIM))[lane];
        float d0 = v.x * w0.x + v.y * w0.y + v.z * w0.z + v.w * w0.w;
        float d1 = v.x * w1.x + v.y * w1.y + v.z * w1.z + v.w * w1.w;
#pragma unroll
        for (int off = 16; off > 0; off >>= 1) {
            d0 += __shfl_xor(d0, off, 32);
            d1 += __shfl_xor(d1, off, 32);
        }
        if (lane == 0) {
            logits[(size_t)node * 2 + 0] = d0 + bc[0];
            logits[(size_t)node * 2 + 1] = d1 + bc[1];
        }
    }
}

// ---------------- log-softmax over node dimension ----------------
// blockIdx.x = class index c; computes lse[c] = max_n + log(sum_n exp(x - max))
__global__ __launch_bounds__(256)
void gs_lse_kernel(const float* __restrict__ logits, float* __restrict__ lse, int n) {
    const int c = blockIdx.x;
    __shared__ float red[256];
    float m = -3.0e38f;
    for (int i = threadIdx.x; i < n; i += 256)
        m = fmaxf(m, logits[(size_t)i * 2 + c]);
    red[threadIdx.x] = m;
    __syncthreads();
    for (int s = 128; s > 0; s >>= 1) {
        if (threadIdx.x < s) red[threadIdx.x] = fmaxf(red[threadIdx.x], red[threadIdx.x + s]);
        __syncthreads();
    }
    const float gm = red[0];
    __syncthreads();
    float ssum = 0.0f;
    for (int i = threadIdx.x; i < n; i += 256)
        ssum += __expf(logits[(size_t)i * 2 + c] - gm);
    red[threadIdx.x] = ssum;
    __syncthreads();
    for (int s = 128; s > 0; s >>= 1) {
        if (threadIdx.x < s) red[threadIdx.x] += red[threadIdx.x + s];
        __syncthreads();
    }
    if (threadIdx.x == 0) lse[c] = gm + __logf(red[0]);
}

__global__ __launch_bounds__(256)
void gs_sub_lse_kernel(float* __restrict__ logits, const float* __restrict__ lse, int n2) {
    int i = blockIdx.x * blockDim.x + threadIdx.x;
    int stride = gridDim.x * blockDim.x;
    for (; i < n2; i += stride) logits[i] -= lse[i & 1];
}

// ---------------- host-side orchestration ----------------
extern "C" void kernel_launch(void* const* d_in, const int* in_sizes, int n_in,
                              void* d_out, int out_size, void* d_ws, size_t ws_size,
                              hipStream_t stream) {
    const int T = 4;
    const int E = in_sizes[0] / (2 * T);        // 800000
    const int N = in_sizes[1] / (T * FDIM);     // 50000

    const int*   graph = (const int*)  d_in[0];
    const float* fts   = (const float*)d_in[1];
    const float* W1l   = (const float*)d_in[3];
    const float* b1    = (const float*)d_in[4];
    const float* W1r   = (const float*)d_in[5];
    const float* W2l   = (const float*)d_in[6];
    const float* b2    = (const float*)d_in[7];
    const float* W2r   = (const float*)d_in[8];
    const float* Wc    = (const float*)d_in[9];
    const float* bc    = (const float*)d_in[10];

    // workspace layout (floats): sum[N*128] | cnt[N] | ys[N*128] | y2[N*128] | lse[2]
    float* sum = (float*)d_ws;
    float* cnt = sum + (size_t)N * FDIM;
    float* ys  = cnt + N;
    float* y2  = ys  + (size_t)N * FDIM;
    float* lse = y2  + (size_t)N * FDIM;
    float* outF = (float*)d_out;

    const int* srcLast = graph + (size_t)(T - 1) * 2 * E;
    const int* dstLast = srcLast + E;

    const int zn = N * FDIM + N;                  // sum + cnt contiguous
    const int zeroGrid    = (zn + 255) / 256;
    const int scatterGrid = 4096;                 // grid-stride over E waves
    const int gemmGrid    = N / 16;               // 3125, exact
    const int clsGrid     = 2048;
    const int subGrid     = (2 * N + 255) / 256;

    for (int t = 0; t < T; ++t) {
        const float* xt  = fts + (size_t)t * N * FDIM;
        const int*   src = graph + (size_t)t * 2 * E;
        const int*   dst = src + E;
        float* logits_t  = outF + (size_t)t * N * 2;

        // ---- layer 1 (graph_t) ----
        gs_zero_kernel<<<zeroGrid, 256, 0, stream>>>(sum, zn);
        gs_scatter_kernel<<<scatterGrid, 256, 0, stream>>>(xt, src, dst, sum, cnt, E);
        gs_sage_gemm_kernel<<<gemmGrid, 256, 0, stream>>>(sum, cnt, xt, W1l, b1, W1r, ys, 1);

        // ---- layer 2 (last graph) ----
        gs_zero_kernel<<<zeroGrid, 256, 0, stream>>>(sum, zn);
        gs_scatter_kernel<<<scatterGrid, 256, 0, stream>>>(ys, srcLast, dstLast, sum, cnt, E);
        gs_sage_gemm_kernel<<<gemmGrid, 256, 0, stream>>>(sum, cnt, ys, W2l, b2, W2r, y2, 1);

        // ---- classifier + log_softmax over nodes ----
        gs_classifier_kernel<<<clsGrid, 256, 0, stream>>>(y2, Wc, bc, logits_t, N);
        gs_lse_kernel<<<2, 256, 0, stream>>>(logits_t, lse, N);
        gs_sub_lse_kernel<<<subGrid, 256, 0, stream>>>(logits_t, lse, 2 * N);
    }
}